// HyenaLayer_38680475467970
// MI455X (gfx1250) — compile-verified
//
#include <hip/hip_runtime.h>

// ---------------------------------------------------------------------------
// Hyena layer for MI455X (gfx1250): WMMA f32 GEMM projections + LDS FFT conv.
// B=8, S=4096, D=256, STAGES=3, L=2S=8192.
// ---------------------------------------------------------------------------

#define BB 8
#define SS 4096
#define DD 256
#define LN 8192          // FFT length (complex points, packed y + i*f)
#define LOG2LN 13

typedef float v2f __attribute__((ext_vector_type(2)));
typedef float v8f __attribute__((ext_vector_type(8)));

typedef __attribute__((address_space(1))) int gint_t;   // global-AS int
typedef __attribute__((address_space(3))) int lint_t;   // LDS-AS int

// ---------------------------------------------------------------------------
// Positional encoding table: pe[s*256 + d] = d<128 ? sin(s*w_d) : cos(s*w_d)
// ---------------------------------------------------------------------------
__global__ void pe_kernel(float* __restrict__ pe) {
    int idx = blockIdx.x * blockDim.x + threadIdx.x;   // [0, S*D)
    int s = idx >> 8;
    int d = idx & 255;
    int i = d & 127;
    float freq = __powf(10000.0f, -(float)i / 128.0f);
    float phi = (float)s * freq;
    pe[idx] = (d < 128) ? __sinf(phi) : __cosf(phi);
}

// ---------------------------------------------------------------------------
// WMMA f32 GEMM:  x[k][b][w][s] = sum_d (X[b][s][d]*mask) * W_proj[d][w][k]
// One 16x16 output tile per wave, K stepped by 4 via v_wmma_f32_16x16x4_f32.
// Output written time-major so the FFT kernel gets contiguous sequences.
// grid = (Mtiles=2048, 2, 4), block = 256 (8 waves)
// ---------------------------------------------------------------------------
__global__ __launch_bounds__(256) void proj_x_kernel(
    const float* __restrict__ X, const int* __restrict__ lengths,
    const float* __restrict__ Wproj, float* __restrict__ xbuf) {
    int lane = threadIdx.x & 31;
    int wave = threadIdx.x >> 5;
    int mtile = blockIdx.x;                  // 0..2047  (M = B*S = 32768)
    int ntile = blockIdx.y * 8 + wave;       // 0..15    (N = 256)
    int kst   = blockIdx.z;                  // 0..3

    int m0 = mtile * 16;
    int n0 = ntile * 16;
    int b  = m0 >> 12;                       // rows of a tile share one batch
    int s0 = m0 & 4095;

    int arow = lane & 15;
    int koff = (lane >> 4) * 2;              // A: K = 2*(lane>>4) + {0,1}
    int s    = s0 + arow;
    float msk = (s < lengths[b]) ? 1.0f : 0.0f;
    const float* aptr = X + ((size_t)(b * SS + s)) * DD + koff;

    int ncol = n0 + (lane & 15);
    int bk   = (lane >> 4) * 2;              // B: K = 2*(lane>>4) + v
    const float* bptr = Wproj + kst;         // element (d,w,k) at ((d*256)+w)*4+k

    v8f c = {};
    for (int kk = 0; kk < DD; kk += 4) {
        __builtin_prefetch(aptr + kk + 32, 0, 1);   // -> global_prefetch_b8
        v2f a = *(const v2f*)(aptr + kk);
        a *= msk;
        v2f bfrag;
        bfrag.x = bptr[((size_t)(kk + bk + 0) * DD + ncol) * 4];
        bfrag.y = bptr[((size_t)(kk + bk + 1) * DD + ncol) * 4];
        c = __builtin_amdgcn_wmma_f32_16x16x4_f32(
                false, a, false, bfrag, (short)0, c, false, false);
    }

    size_t base = ((size_t)(kst * BB + b) * DD + ncol) * SS;
    int mhi = (lane >> 4) << 3;
    for (int v = 0; v < 8; ++v) {
        xbuf[base + (size_t)(s0 + v + mhi)] = c[v];
    }
}

// ---------------------------------------------------------------------------
// WMMA f32 GEMM:  f[k][w][s] = sum_d PE[s][d] * W_filt[d][w][k]
// grid = (Mtiles=256, 2, 3), block = 256
// ---------------------------------------------------------------------------
__global__ __launch_bounds__(256) void proj_f_kernel(
    const float* __restrict__ pe, const float* __restrict__ Wfilt,
    float* __restrict__ fbuf) {
    int lane = threadIdx.x & 31;
    int wave = threadIdx.x >> 5;
    int mtile = blockIdx.x;                  // 0..255 (M = S = 4096)
    int ntile = blockIdx.y * 8 + wave;       // 0..15
    int kst   = blockIdx.z;                  // 0..2

    int s0 = mtile * 16;
    int n0 = ntile * 16;

    int arow = lane & 15;
    int koff = (lane >> 4) * 2;
    const float* aptr = pe + ((size_t)(s0 + arow)) * DD + koff;

    int ncol = n0 + (lane & 15);
    int bk   = (lane >> 4) * 2;
    const float* bptr = Wfilt + kst;         // element (d,w,k) at ((d*256)+w)*3+k

    v8f c = {};
    for (int kk = 0; kk < DD; kk += 4) {
        __builtin_prefetch(aptr + kk + 32, 0, 1);
        v2f a = *(const v2f*)(aptr + kk);
        v2f bfrag;
        bfrag.x = bptr[((size_t)(kk + bk + 0) * DD + ncol) * 3];
        bfrag.y = bptr[((size_t)(kk + bk + 1) * DD + ncol) * 3];
        c = __builtin_amdgcn_wmma_f32_16x16x4_f32(
                false, a, false, bfrag, (short)0, c, false, false);
    }

    size_t base = (size_t)kst * (DD * SS) + (size_t)ncol * SS;
    int mhi = (lane >> 4) << 3;
    for (int v = 0; v < 8; ++v) {
        fbuf[base + (size_t)(s0 + v + mhi)] = c[v];
    }
}

// ---------------------------------------------------------------------------
// FFT helpers (LDS-resident, 512 threads, length-8192 radix-2 DIF).
// DIF forward leaves bit-reversed order; a bit-reversal pass restores it.
// Inverse uses the conjugation trick (conjugate spectrum, forward FFT, /N).
// ---------------------------------------------------------------------------
__device__ inline void fft_dif(float2* z, int tid) {
    for (int st = 0; st < LOG2LN; ++st) {
        int lh = (LOG2LN - 1) - st;          // log2(half)
        int h  = 1 << lh;
        for (int r = 0; r < 8; ++r) {
            int p = tid + 512 * r;           // 0..4095 butterflies
            int j = p & (h - 1);
            int base = ((p >> lh) << (lh + 1)) + j;
            float2 u = z[base];
            float2 v = z[base + h];
            float2 sum  = make_float2(u.x + v.x, u.y + v.y);
            float2 diff = make_float2(u.x - v.x, u.y - v.y);
            float ang = -3.14159265358979323846f * (float)j / (float)h;
            float sn, cs;
            __sincosf(ang, &sn, &cs);
            z[base]     = sum;
            z[base + h] = make_float2(diff.x * cs - diff.y * sn,
                                      diff.x * sn + diff.y * cs);
        }
        __syncthreads();
    }
}

__device__ inline void bitrev_pass(float2* z, int tid) {
    for (int r = 0; r < 16; ++r) {
        int i = tid + 512 * r;
        int rv = (int)(__brev((unsigned)i) >> (32 - LOG2LN));
        if (rv > i) {
            float2 t = z[i];
            z[i] = z[rv];
            z[rv] = t;
        }
    }
    __syncthreads();
}

// Async fill of the packed complex LDS array via GLOBAL_LOAD_ASYNC_TO_LDS_B32:
// y[n] -> z[n].x, f[n] -> z[n].y (EXEC-predicated by the ragged mask; LDS was
// pre-zeroed, which also provides the causal zero pad).
__device__ inline void async_fill(float2* z, const float* yrow,
                                  const float* frow, int len, int tid) {
    for (int r = 0; r < 8; ++r) {
        int n = tid + 512 * r;              // 0..4095
#if __has_builtin(__builtin_amdgcn_global_load_async_to_lds_b32)
        __builtin_amdgcn_global_load_async_to_lds_b32(
            (gint_t*)(yrow + n), (lint_t*)&z[n].x, 0, 0);
        if (n < len) {
            __builtin_amdgcn_global_load_async_to_lds_b32(
                (gint_t*)(frow + n), (lint_t*)&z[n].y, 0, 0);
        }
#else
        {
            unsigned lds_x = (unsigned)(size_t)(lint_t*)&z[n].x;
            const float* gy = yrow + n;
            asm volatile("global_load_async_to_lds_b32 %0, %1, off"
                         :: "v"(lds_x), "v"(gy) : "memory");
            if (n < len) {
                unsigned lds_y = (unsigned)(size_t)(lint_t*)&z[n].y;
                const float* gf = frow + n;
                asm volatile("global_load_async_to_lds_b32 %0, %1, off"
                             :: "v"(lds_y), "v"(gf) : "memory");
            }
        }
#endif
    }
#if __has_builtin(__builtin_amdgcn_s_wait_asynccnt)
    __builtin_amdgcn_s_wait_asynccnt(0);
#else
    asm volatile("s_wait_asynccnt 0" ::: "memory");
#endif
}

// ---------------------------------------------------------------------------
// One Hyena FFT-conv stage for one (b,d) sequence:
//   y_out = irfft( rfft(pad(y_in)) * rfft(pad(f*mask)) ) * gate
// Packed-real trick: one complex FFT of (y + i*f) yields both spectra.
// grid = B*D = 2048 blocks, 512 threads, 64 KB LDS.
// ---------------------------------------------------------------------------
__global__ __launch_bounds__(512) void fftconv_kernel(
    const float* __restrict__ yin,      // [B][D][S] time-major
    const float* __restrict__ fstage,   // [D][S]
    const float* __restrict__ gate,     // [B][D][S] (already masked)
    const int* __restrict__ lengths,
    float* __restrict__ yout,
    int out_transposed)                 // 1 => write [B][S][D] (d_out layout)
{
    __shared__ float2 z[LN];            // 64 KB of the 320 KB WGP LDS
    int tid = threadIdx.x;
    int b = blockIdx.x >> 8;
    int d = blockIdx.x & 255;
    int len = lengths[b];

    const float* yrow = yin + ((size_t)(b * DD + d)) * SS;
    const float* frow = fstage + (size_t)d * SS;

    // Pre-zero LDS: gives the causal pad [S,2S) and the masked tail of f.
    for (int r = 0; r < 8; ++r) {
        int n = tid + 512 * r;
        z[n]      = make_float2(0.0f, 0.0f);
        z[n + SS] = make_float2(0.0f, 0.0f);
    }
    __syncthreads();                    // drain DS stores before async writes

    async_fill(z, yrow, frow, len, tid);
    __syncthreads();

    // forward FFT of packed signal
    fft_dif(z, tid);
    bitrev_pass(z, tid);

    // Hermitian unpack, spectral multiply, store conj(P) for inverse trick.
    for (int r = 0; r < 8; ++r) {
        int k = tid + 512 * r;          // 0..4095
        if (k == 0) {
            float2 z0 = z[0];
            z[0] = make_float2(z0.x * z0.y, 0.0f);          // DC: Fy=Re, Ff=Im
            float2 zh = z[SS];
            z[SS] = make_float2(zh.x * zh.y, 0.0f);         // Nyquist
        } else {
            float2 a  = z[k];
            float2 bm = z[LN - k];
            float2 Fy = make_float2(0.5f * (a.x + bm.x), 0.5f * (a.y - bm.y));
            float2 Ff = make_float2(0.5f * (a.y + bm.y), -0.5f * (a.x - bm.x));
            float2 P  = make_float2(Fy.x * Ff.x - Fy.y * Ff.y,
                                    Fy.x * Ff.y + Fy.y * Ff.x);
            z[k]      = make_float2(P.x, -P.y);   // conj(P[k])
            z[LN - k] = P;                        // conj(P[N-k]) = P[k]
        }
    }
    __syncthreads();

    // inverse FFT via conjugation trick: ifft(P) = conj(fft(conj(P)))/N
    fft_dif(z, tid);
    bitrev_pass(z, tid);

    const float scale = 1.0f / (float)LN;
    const float* grow = gate + ((size_t)(b * DD + d)) * SS;
    for (int r = 0; r < 8; ++r) {
        int n = tid + 512 * r;          // only first S samples survive gating
        float val = z[n].x * scale * grow[n];
        if (out_transposed)
            yout[((size_t)(b * SS + n)) * DD + d] = val;
        else
            yout[((size_t)(b * DD + d)) * SS + n] = val;
    }
}

// ---------------------------------------------------------------------------
// Host-side orchestration (all launches on `stream`; graph-capture safe).
// Workspace layout (floats): pe[1M] | xbuf[4*8M] | fbuf[3*1M] | ybuf[8M]
//   total = 46,137,344 floats = 176 MB
// ---------------------------------------------------------------------------
extern "C" void kernel_launch(void* const* d_in, const int* in_sizes, int n_in,
                              void* d_out, int out_size, void* d_ws, size_t ws_size,
                              hipStream_t stream) {
    const float* X       = (const float*)d_in[0];
    const int*   lengths = (const int*)d_in[1];
    const float* Wproj   = (const float*)d_in[2];
    const float* Wfilt   = (const float*)d_in[3];
    float* out = (float*)d_out;

    const size_t PE_N   = (size_t)SS * DD;            // 1,048,576
    const size_t XSLAB  = (size_t)BB * DD * SS;       // 8,388,608 per k-stage
    const size_t FSLAB  = (size_t)DD * SS;            // 1,048,576 per stage

    float* ws   = (float*)d_ws;
    float* pe   = ws;
    float* xbuf = pe + PE_N;
    float* fbuf = xbuf + 4 * XSLAB;
    float* ybuf = fbuf + 3 * FSLAB;

    // 1) positional-encoding table
    pe_kernel<<<(SS * DD) / 256, 256, 0, stream>>>(pe);

    // 2) WMMA projections: x (masked) and f, both written time-major
    proj_x_kernel<<<dim3(2048, 2, 4), 256, 0, stream>>>(X, lengths, Wproj, xbuf);
    proj_f_kernel<<<dim3(256, 2, 3), 256, 0, stream>>>(pe, Wfilt, fbuf);

    // 3) three FFT-conv stages; y0 = x[...,0] lives in xbuf slab 0
    fftconv_kernel<<<BB * DD, 512, 0, stream>>>(
        xbuf + 0 * XSLAB, fbuf + 0 * FSLAB, xbuf + 1 * XSLAB, lengths, ybuf, 0);
    fftconv_kernel<<<BB * DD, 512, 0, stream>>>(
        ybuf, fbuf + 1 * FSLAB, xbuf + 2 * XSLAB, lengths, ybuf, 0);
    fftconv_kernel<<<BB * DD, 512, 0, stream>>>(
        ybuf, fbuf + 2 * FSLAB, xbuf + 3 * XSLAB, lengths, out, 1);
}